// MessagePassingNet_65841848648262
// MI455X (gfx1250) — compile-verified
//
#include <hip/hip_runtime.h>
#include <hip/hip_bf16.h>

#define H      128
#define NG     256
#define SLOTS  64
#define LN_EPS 1e-5f

typedef float v2f __attribute__((ext_vector_type(2)));
typedef float v8f __attribute__((ext_vector_type(8)));

__device__ __forceinline__ float wave_sum32(float v) {
#pragma unroll
  for (int off = 16; off > 0; off >>= 1) v += __shfl_xor(v, off, 32);
  return v;
}

// ---------------- zero scratch accumulators ----------------
__global__ void k_zero(float* __restrict__ p, long n) {
  long i = (long)blockIdx.x * blockDim.x + threadIdx.x;
  long st = (long)gridDim.x * blockDim.x;
  for (; i < n; i += st) p[i] = 0.f;
}

// ---------------- edge / graph statistics (once) ----------------
__global__ void k_edge_stats(const int* __restrict__ ei, const float* __restrict__ eattr,
                             const int* __restrict__ batch,
                             float* __restrict__ csrc, float* __restrict__ cdst,
                             float* __restrict__ attr6, float* __restrict__ gcnt,
                             int N, int E) {
  __shared__ float s6[6];
  int tid = threadIdx.x;
  if (tid < 6) s6[tid] = 0.f;
  __syncthreads();
  float part[6] = {0.f, 0.f, 0.f, 0.f, 0.f, 0.f};
  int stride = gridDim.x * blockDim.x;
  for (int e = blockIdx.x * blockDim.x + tid; e < E; e += stride) {
    atomicAdd(&csrc[ei[e]], 1.f);
    atomicAdd(&cdst[ei[E + e]], 1.f);
    const float* a = eattr + (long)e * 6;
#pragma unroll
    for (int j = 0; j < 6; ++j) part[j] += a[j];
  }
#pragma unroll
  for (int j = 0; j < 6; ++j) atomicAdd(&s6[j], part[j]);
  for (int n = blockIdx.x * blockDim.x + tid; n < N; n += stride)
    atomicAdd(&gcnt[batch[n]], 1.f);
  __syncthreads();
  if (tid < 6) atomicAdd(&attr6[tid], s6[tid]);
}

// ---------------- h = x @ w_node + b_node  (WMMA f32), fused layer-0 sums ----------------
__global__ void __launch_bounds__(256) k_node_embed(
    const float* __restrict__ x, const float* __restrict__ w_node,
    const float* __restrict__ b_node, const float* __restrict__ csrc,
    const float* __restrict__ cdst, float* __restrict__ h,
    float* __restrict__ sums0, int N) {
  __shared__ float xs[16 * 66];
  __shared__ float cs[16], cd[16];
  int tid = threadIdx.x;
  int rowbase = blockIdx.x * 16;
  for (int t = tid; t < 16 * 66; t += 256) {
    int r = t / 66, c = t % 66;
    int gr = rowbase + r;
    xs[t] = (gr < N) ? x[(long)gr * 66 + c] : 0.f;
  }
  if (tid < 16) {
    int gr = rowbase + tid;
    cs[tid] = (gr < N) ? csrc[gr] : 0.f;
    cd[tid] = (gr < N) ? cdst[gr] : 0.f;
  }
  __syncthreads();

  int wave = tid >> 5, lane = tid & 31;
  int mA   = lane & 15;
  int koff = (lane >> 4) * 2;
  int n    = wave * 16 + (lane & 15);

  v8f acc = {0.f, 0.f, 0.f, 0.f, 0.f, 0.f, 0.f, 0.f};
#pragma unroll
  for (int kk = 0; kk < 64; kk += 4) {
    v2f a, b;
    a.x = xs[mA * 66 + kk + koff];
    a.y = xs[mA * 66 + kk + koff + 1];
    b.x = w_node[(kk + koff) * H + n];
    b.y = w_node[(kk + koff + 1) * H + n];
    acc = __builtin_amdgcn_wmma_f32_16x16x4_f32(false, a, false, b, (short)0, acc,
                                                false, false);
  }
  float w64 = w_node[64 * H + n];
  float w65 = w_node[65 * H + n];
  float bn  = b_node[n];
  float psrc = 0.f, pdst = 0.f;
#pragma unroll
  for (int v = 0; v < 8; ++v) {
    int m = v + 8 * (lane >> 4);
    float hv = acc[v] + xs[m * 66 + 64] * w64 + xs[m * 66 + 65] * w65 + bn;
    int gr = rowbase + m;
    if (gr < N) h[(long)gr * H + n] = hv;
    psrc += cs[m] * hv;
    pdst += cd[m] * hv;
  }
  int slot = blockIdx.x & (SLOTS - 1);
  atomicAdd(&sums0[slot * (2 * H) + n], psrc);
  atomicAdd(&sums0[slot * (2 * H) + H + n], pdst);
}

// ---------------- per-layer: mean_msg from accumulated sums ----------------
__global__ void __launch_bounds__(128) k_finalize(
    const float* __restrict__ sums, const float* __restrict__ attr6,
    const float* __restrict__ w_edge, const float* __restrict__ b_edge,
    const float* __restrict__ mpw, const float* __restrict__ mpb,
    float* __restrict__ mean_msg, float Ef) {
  __shared__ float Ss[H], Sd[H], Se[H];
  int j = threadIdx.x;
  float a = 0.f, b = 0.f;
  for (int s = 0; s < SLOTS; ++s) {
    a += sums[s * (2 * H) + j];
    b += sums[s * (2 * H) + H + j];
  }
  Ss[j] = a;
  Sd[j] = b;
  float ef = Ef * b_edge[j];
#pragma unroll
  for (int k = 0; k < 6; ++k) ef += attr6[k] * w_edge[k * H + j];
  Se[j] = ef;
  __syncthreads();
  float acc = 0.f;
  for (int k = 0; k < H; ++k) {
    acc += Ss[k] * mpw[k * H + j];
    acc += Sd[k] * mpw[(H + k) * H + j];
    acc += Se[k] * mpw[(2 * H + k) * H + j];
  }
  mean_msg[j] = acc / Ef + mpb[j];
}

// ---------------- per-layer: h += mean_msg; LN; ReLU; next sums or pooling ----------------
__global__ void __launch_bounds__(256) k_update(
    float* __restrict__ h, const float* __restrict__ mean_msg,
    const float* __restrict__ lng, const float* __restrict__ lnb,
    const float* __restrict__ csrc, const float* __restrict__ cdst,
    const int* __restrict__ batch, float* __restrict__ sums_next,
    float* __restrict__ pooled, int N, int last, int npw) {
  __shared__ float mm[H];
  int tid = threadIdx.x;
  if (tid < H) mm[tid] = mean_msg[tid];
  __syncthreads();

  int wave = blockIdx.x * 8 + (tid >> 5);
  int lane = tid & 31;
  int c0   = lane * 4;
  int start = wave * npw;
  int end   = min(N, start + npw);

  float m0 = mm[c0], m1 = mm[c0 + 1], m2 = mm[c0 + 2], m3 = mm[c0 + 3];
  float g0 = lng[c0], g1 = lng[c0 + 1], g2 = lng[c0 + 2], g3 = lng[c0 + 3];
  float t0 = lnb[c0], t1 = lnb[c0 + 1], t2 = lnb[c0 + 2], t3 = lnb[c0 + 3];

  float as0 = 0, as1 = 0, as2 = 0, as3 = 0;
  float ad0 = 0, ad1 = 0, ad2 = 0, ad3 = 0;
  int curg = -1;
  float p0 = 0, p1 = 0, p2 = 0, p3 = 0;

  for (int n = start; n < end; ++n) {
    float4 v = *(const float4*)(h + (long)n * H + c0);
    v.x += m0; v.y += m1; v.z += m2; v.w += m3;
    float s  = wave_sum32(v.x + v.y + v.z + v.w);
    float mu = s * (1.f / H);
    float dx = v.x - mu, dy = v.y - mu, dz = v.z - mu, dw = v.w - mu;
    float q    = wave_sum32(dx * dx + dy * dy + dz * dz + dw * dw);
    float rstd = rsqrtf(q * (1.f / H) + LN_EPS);
    float y0 = fmaxf(0.f, dx * rstd * g0 + t0);
    float y1 = fmaxf(0.f, dy * rstd * g1 + t1);
    float y2 = fmaxf(0.f, dz * rstd * g2 + t2);
    float y3 = fmaxf(0.f, dw * rstd * g3 + t3);
    float4 yv; yv.x = y0; yv.y = y1; yv.z = y2; yv.w = y3;
    *(float4*)(h + (long)n * H + c0) = yv;

    if (!last) {
      float csn = csrc[n], cdn = cdst[n];
      as0 += csn * y0; as1 += csn * y1; as2 += csn * y2; as3 += csn * y3;
      ad0 += cdn * y0; ad1 += cdn * y1; ad2 += cdn * y2; ad3 += cdn * y3;
    } else {
      int g = batch[n];
      if (g != curg) {
        if (curg >= 0) {
          atomicAdd(&pooled[curg * H + c0 + 0], p0);
          atomicAdd(&pooled[curg * H + c0 + 1], p1);
          atomicAdd(&pooled[curg * H + c0 + 2], p2);
          atomicAdd(&pooled[curg * H + c0 + 3], p3);
        }
        curg = g;
        p0 = p1 = p2 = p3 = 0.f;
      }
      p0 += y0; p1 += y1; p2 += y2; p3 += y3;
    }
  }

  if (!last) {
    int slot = wave & (SLOTS - 1);
    float* sp = sums_next + slot * (2 * H);
    atomicAdd(&sp[c0 + 0], as0); atomicAdd(&sp[c0 + 1], as1);
    atomicAdd(&sp[c0 + 2], as2); atomicAdd(&sp[c0 + 3], as3);
    atomicAdd(&sp[H + c0 + 0], ad0); atomicAdd(&sp[H + c0 + 1], ad1);
    atomicAdd(&sp[H + c0 + 2], ad2); atomicAdd(&sp[H + c0 + 3], ad3);
  } else if (curg >= 0) {
    atomicAdd(&pooled[curg * H + c0 + 0], p0);
    atomicAdd(&pooled[curg * H + c0 + 1], p1);
    atomicAdd(&pooled[curg * H + c0 + 2], p2);
    atomicAdd(&pooled[curg * H + c0 + 3], p3);
  }
}

// ---------------- out = (pooled / cnt) @ w_out + b_out  (WMMA f32) ----------------
__global__ void __launch_bounds__(32) k_out(
    const float* __restrict__ pooled, const float* __restrict__ gcnt,
    const float* __restrict__ w_out, const float* __restrict__ b_out,
    float* __restrict__ out) {
  int lane = threadIdx.x;
  int m0 = (blockIdx.x >> 3) * 16;
  int n0 = (blockIdx.x & 7) * 16;
  int mA   = lane & 15;
  int koff = (lane >> 4) * 2;
  int n    = n0 + (lane & 15);

  float rA = 1.f / fmaxf(gcnt[m0 + mA], 1.f);
  const float* prow = pooled + (long)(m0 + mA) * H;

  v8f acc = {0.f, 0.f, 0.f, 0.f, 0.f, 0.f, 0.f, 0.f};
#pragma unroll 4
  for (int k = 0; k < H; k += 4) {
    v2f a, b;
    a.x = prow[k + koff] * rA;
    a.y = prow[k + koff + 1] * rA;
    b.x = w_out[(k + koff) * H + n];
    b.y = w_out[(k + koff + 1) * H + n];
    acc = __builtin_amdgcn_wmma_f32_16x16x4_f32(false, a, false, b, (short)0, acc,
                                                false, false);
  }
  float bo = b_out[n];
#pragma unroll
  for (int v = 0; v < 8; ++v) {
    int m = v + 8 * (lane >> 4);
    out[(long)(m0 + m) * H + n] = acc[v] + bo;
  }
}

// ---------------- driver ----------------
extern "C" void kernel_launch(void* const* d_in, const int* in_sizes, int n_in,
                              void* d_out, int out_size, void* d_ws, size_t ws_size,
                              hipStream_t stream) {
  const float* x      = (const float*)d_in[0];
  const int*   ei     = (const int*)d_in[1];
  const float* eattr  = (const float*)d_in[2];
  const int*   batch  = (const int*)d_in[3];
  const float* w_node = (const float*)d_in[4];
  const float* b_node = (const float*)d_in[5];
  const float* w_edge = (const float*)d_in[6];
  const float* b_edge = (const float*)d_in[7];
  const float* mpw    = (const float*)d_in[8];
  const float* mpb    = (const float*)d_in[9];
  const float* lng    = (const float*)d_in[10];
  const float* lnb    = (const float*)d_in[11];
  const float* w_out  = (const float*)d_in[12];
  const float* b_out  = (const float*)d_in[13];
  float* out = (float*)d_out;

  int N = in_sizes[0] / 66;
  int E = in_sizes[2] / 6;

  float* ws = (float*)d_ws;
  long off = 0;
  float* h        = ws + off; off += (long)N * H;
  float* csrc     = ws + off; off += N;
  float* cdst     = ws + off; off += N;
  float* attr6    = ws + off; off += 8;
  float* gcnt     = ws + off; off += NG;
  float* mean_msg = ws + off; off += H;
  float* sums     = ws + off; off += 3L * SLOTS * 2 * H;
  float* pooled   = ws + off; off += (long)NG * H;

  long zlen = off - (long)N * H;  // everything after h needs zeroing each call
  hipLaunchKernelGGL(k_zero, dim3(256), dim3(256), 0, stream, csrc, zlen);
  hipLaunchKernelGGL(k_edge_stats, dim3(512), dim3(256), 0, stream,
                     ei, eattr, batch, csrc, cdst, attr6, gcnt, N, E);

  int mtiles = (N + 15) / 16;
  hipLaunchKernelGGL(k_node_embed, dim3(mtiles), dim3(256), 0, stream,
                     x, w_node, b_node, csrc, cdst, h, sums, N);

  int nwaves = 64 * 8;
  int npw = (N + nwaves - 1) / nwaves;
  for (int i = 0; i < 3; ++i) {
    hipLaunchKernelGGL(k_finalize, dim3(1), dim3(128), 0, stream,
                       sums + (long)i * SLOTS * 2 * H, attr6, w_edge, b_edge,
                       mpw + (long)i * 3 * H * H, mpb + (long)i * H,
                       mean_msg, (float)E);
    int last = (i == 2);
    float* sums_next = sums + (long)((i + 1 < 3) ? i + 1 : 0) * SLOTS * 2 * H;
    hipLaunchKernelGGL(k_update, dim3(64), dim3(256), 0, stream,
                       h, mean_msg, lng, lnb, csrc, cdst, batch,
                       sums_next, pooled, N, last, npw);
  }

  hipLaunchKernelGGL(k_out, dim3(128), dim3(32), 0, stream,
                     pooled, gcnt, w_out, b_out, out);
}